// voltageNN_1477468750213
// MI455X (gfx1250) — compile-verified
//
#include <hip/hip_runtime.h>
#include <math.h>

#define HD 256
#define TT 1000
#define LL 5
#define BB 128

typedef float v2f __attribute__((ext_vector_type(2)));
typedef float v8f __attribute__((ext_vector_type(8)));

__device__ __forceinline__ float sigf(float v) {
    return 1.0f / (1.0f + expf(-v));
}

// ---------------------------------------------------------------------------
// Kernel 1: full 5-layer projected LSTM scan. One wave32 per batch element.
// Each lane owns 8 hidden units; per-timestep reduction is an intra-wave
// shfl_xor butterfly (no barriers on the recurrence critical path).
// ---------------------------------------------------------------------------
__global__ __launch_bounds__(32) void lstm_scan_kernel(
    const float* __restrict__ x,    // [B,T,1]
    const float* __restrict__ Wih,  // [L,4H,1]
    const float* __restrict__ Whh,  // [L,4H,1]
    const float* __restrict__ bih,  // [L,4H]
    const float* __restrict__ bhh,  // [L,4H]
    const float* __restrict__ Whr,  // [L,1,H]
    float* __restrict__ seq_out)    // [B,T]
{
    __shared__ float bufA[TT];
    __shared__ float bufB[TT];

    const int b    = blockIdx.x;
    const int lane = threadIdx.x;

    // load x[b, :, 0] into bufA
    for (int t = lane; t < TT; t += 32) bufA[t] = x[b * TT + t];
    __syncthreads();

    float* inb  = bufA;
    float* outb = bufB;

    for (int l = 0; l < LL; ++l) {
        const int base = l * 4 * HD;
        float wi[4][8], wh[4][8], bs[4][8], wr[8], c[8];
#pragma unroll
        for (int j = 0; j < 8; ++j) {
            const int hidx = lane * 8 + j;
#pragma unroll
            for (int g = 0; g < 4; ++g) {
                wi[g][j] = Wih[base + g * HD + hidx];
                wh[g][j] = Whh[base + g * HD + hidx];
                bs[g][j] = bih[base + g * HD + hidx] + bhh[base + g * HD + hidx];
            }
            wr[j] = Whr[l * HD + hidx];
            c[j]  = 0.0f;
        }

        float hprev = 0.0f;
        for (int t = 0; t < TT; ++t) {
            const float xv = inb[t];   // uniform LDS broadcast
            float part = 0.0f;
#pragma unroll
            for (int j = 0; j < 8; ++j) {
                float gi = sigf(  fmaf(xv, wi[0][j], fmaf(hprev, wh[0][j], bs[0][j])));
                float gf = sigf(  fmaf(xv, wi[1][j], fmaf(hprev, wh[1][j], bs[1][j])));
                float gg = tanhf( fmaf(xv, wi[2][j], fmaf(hprev, wh[2][j], bs[2][j])));
                float go = sigf(  fmaf(xv, wi[3][j], fmaf(hprev, wh[3][j], bs[3][j])));
                float cn = fmaf(gf, c[j], gi * gg);
                c[j] = cn;
                part = fmaf(go * tanhf(cn), wr[j], part);
            }
            // intra-wave butterfly: all 32 lanes end with the full sum
#pragma unroll
            for (int off = 16; off > 0; off >>= 1)
                part += __shfl_xor(part, off, 32);
            hprev = part;
            if (lane == 0) outb[t] = part;
        }
        __syncthreads();
        float* tmp = inb; inb = outb; outb = tmp;
    }

    // after the swap, the last layer's output sits in inb
    for (int t = lane; t < TT; t += 32) seq_out[b * TT + t] = inb[t];
}

// ---------------------------------------------------------------------------
// Kernel 2: f32 WMMA fully-connected layer: Out = act(A @ W^T + bias)
// A: [128,K] row-major, W: [N,K] row-major. One wave per 16x16 output tile,
// K-loop in steps of 4 using V_WMMA_F32_16X16X4_F32.
// A-frag (16x4): lanes 0-15 -> {K=k,k+1}, lanes 16-31 -> {K=k+2,k+3}.
// B-frag (4x16) mirrored on the N index. C/D: VGPR r, lanes 0-15 -> M=r,
// lanes 16-31 -> M=r+8.
//
// Padding lanes (bcol >= N) clamp their row pointer to a valid row and load
// unconditionally: a B column only feeds its own D column, and columns >= N
// are never stored, so no masking / exec divergence is needed.
// act: 0 = sigmoid, 1 = relu
// ---------------------------------------------------------------------------
__global__ __launch_bounds__(32) void fc_wmma_kernel(
    const float* __restrict__ A,
    const float* __restrict__ W,
    const float* __restrict__ bias,
    float* __restrict__ Out,
    int K, int N, int act)
{
    const int mt   = blockIdx.x;       // row tile (M=128 -> 8 tiles)
    const int nt   = blockIdx.y;       // col tile
    const int lane = threadIdx.x;
    const int half = lane >> 4;        // 0: lanes 0-15, 1: lanes 16-31
    const int l15  = lane & 15;

    const int arow = mt * 16 + l15;
    const int bcol = nt * 16 + l15;
    const int brow = (bcol < N) ? bcol : (N - 1);   // clamp, no predication

    const float* ap = A + (size_t)arow * K;
    const float* bp = W + (size_t)brow * K;

    v8f acc = {};
    for (int k = 0; k < K; k += 4) {
        const int ka = k + half * 2;                // even -> 8B aligned
        v2f av = *(const v2f*)(ap + ka);
        v2f bv = *(const v2f*)(bp + ka);
        acc = __builtin_amdgcn_wmma_f32_16x16x4_f32(
            /*neg_a=*/false, av, /*neg_b=*/false, bv,
            /*c_mod=*/(short)0, acc, /*reuse_a=*/false, /*reuse_b=*/false);
    }

    if (bcol < N) {
        const float bb = bias[bcol];
#pragma unroll
        for (int r = 0; r < 8; ++r) {
            const int row = mt * 16 + r + half * 8;
            float v = acc[r] + bb;
            if (act == 0)      v = 1.0f / (1.0f + expf(-v));
            else               v = fmaxf(v, 0.0f);
            Out[row * N + bcol] = v;
        }
    }
}

// ---------------------------------------------------------------------------
// Kernel 3: final [128,100] @ [100,1] matvec + bias
// ---------------------------------------------------------------------------
__global__ void fc_final_kernel(
    const float* __restrict__ A,   // [128,100]
    const float* __restrict__ W4,  // [1,100]
    const float* __restrict__ b4,  // [1]
    float* __restrict__ Out)       // [128]
{
    const int b = blockIdx.x * blockDim.x + threadIdx.x;
    if (b < BB) {
        float s = b4[0];
#pragma unroll 4
        for (int j = 0; j < 100; ++j) s = fmaf(A[b * 100 + j], W4[j], s);
        Out[b] = s;
    }
}

extern "C" void kernel_launch(void* const* d_in, const int* in_sizes, int n_in,
                              void* d_out, int out_size, void* d_ws, size_t ws_size,
                              hipStream_t stream) {
    const float* x   = (const float*)d_in[0];
    const float* Wih = (const float*)d_in[1];
    const float* Whh = (const float*)d_in[2];
    const float* bih = (const float*)d_in[3];
    const float* bhh = (const float*)d_in[4];
    const float* Whr = (const float*)d_in[5];
    const float* W1  = (const float*)d_in[6];
    const float* b1  = (const float*)d_in[7];
    const float* W2  = (const float*)d_in[8];
    const float* b2  = (const float*)d_in[9];
    const float* W3  = (const float*)d_in[10];
    const float* b3  = (const float*)d_in[11];
    const float* W4  = (const float*)d_in[12];
    const float* b4  = (const float*)d_in[13];

    char* ws = (char*)d_ws;
    float* seqF = (float*)(ws);                              // 128*1000 f32
    float* o1   = (float*)(ws + (size_t)BB * TT * 4);        // 128*100
    float* o2   = (float*)(ws + (size_t)BB * TT * 4 + 51200);
    float* o3   = (float*)(ws + (size_t)BB * TT * 4 + 102400);
    float* out  = (float*)d_out;                             // [128,1]

    lstm_scan_kernel<<<BB, 32, 0, stream>>>(x, Wih, Whh, bih, bhh, Whr, seqF);

    fc_wmma_kernel<<<dim3(8, 7), 32, 0, stream>>>(seqF, W1, b1, o1, 1000, 100, 0);
    fc_wmma_kernel<<<dim3(8, 7), 32, 0, stream>>>(o1,   W2, b2, o2, 100,  100, 0);
    fc_wmma_kernel<<<dim3(8, 7), 32, 0, stream>>>(o2,   W3, b3, o3, 100,  100, 1);

    fc_final_kernel<<<1, BB, 0, stream>>>(o3, W4, b4, out);
}